// MultiheadSelfAttention_66090956751324
// MI455X (gfx1250) — compile-verified
//
#include <hip/hip_runtime.h>

// MI455X / gfx1250, wave32. fp32 WMMA (V_WMMA_F32_16X16X4_F32) everywhere:
// inputs are fp32 and the problem is matrix-core bound (~13 GFLOP vs ~25MB
// of operands @ 23.3 TB/s HBM; everything lives in the 192MB L2 anyway).
// Projection GEMMs use a 16x64 register tile per wave: each A fragment is
// reused across 4 B fragments -> 4 wmma per 5 b64 loads (6.4 FLOP/B).

typedef float v2f __attribute__((ext_vector_type(2)));
typedef float v8f __attribute__((ext_vector_type(8)));

constexpr int Tn = 1024;    // sequence length
constexpr int Dn = 1024;    // model dim
constexpr int NH = 16;      // num_heads
constexpr int HD = Dn / NH; // head_dim = 64 -> 64 attention maps, inner dim 16

// fp32 WMMA helper: D = A(16x4) * B(4x16) + C(16x16)
__device__ __forceinline__ v8f wmma_f32(v2f a, v2f b, v8f c) {
    return __builtin_amdgcn_wmma_f32_16x16x4_f32(false, a, false, b, (short)0, c,
                                                 false, false);
}

// ---------------------------------------------------------------------------
// Kernel 1: Q/K/V projections.  P = X @ W^T + b   (C[m,n] = sum_k X[m,k]W[n,k])
// 256 threads = 8 waves; each wave owns a 16(M) x 64(N) register tile
// (4 accumulators), block covers 32M x 256N.
// A frag: lane m=l&15, half hh=l>>4 -> {X[m, k0+2hh], X[m, k0+2hh+1]}  (b64)
// B frag (B^T case): {W[n, k0+2hh], W[n, k0+2hh+1]}                    (b64)
// ---------------------------------------------------------------------------
__global__ __launch_bounds__(256)
void qkv_proj_kernel(const float* __restrict__ x,
                     const float* __restrict__ wq, const float* __restrict__ bq,
                     const float* __restrict__ wk, const float* __restrict__ bk,
                     const float* __restrict__ wv, const float* __restrict__ bv,
                     float* __restrict__ Qo, float* __restrict__ Ko,
                     float* __restrict__ Vo) {
    const int lane = threadIdx.x & 31;
    const int wave = threadIdx.x >> 5;
    const int z = blockIdx.z;
    const float* W  = (z == 0) ? wq : (z == 1) ? wk : wv;
    const float* Bv = (z == 0) ? bq : (z == 1) ? bk : bv;
    float* Out      = (z == 0) ? Qo : (z == 1) ? Ko : Vo;

    const int m0 = (blockIdx.x * 2 + (wave & 1)) * 16;
    const int n0 = blockIdx.y * 256 + (wave >> 1) * 64;
    const int ml = lane & 15;
    const int hh = lane >> 4;

    const float* arow = x + (m0 + ml) * Dn + 2 * hh;
    const float* brow[4];
#pragma unroll
    for (int j = 0; j < 4; ++j)
        brow[j] = W + (n0 + j * 16 + ml) * Dn + 2 * hh;

    v8f acc[4] = {};
#pragma unroll 2
    for (int k0 = 0; k0 < Dn; k0 += 4) {
        __builtin_prefetch(arow + k0 + 128, 0, 3);
        v2f a = *(const v2f*)(arow + k0);
#pragma unroll
        for (int j = 0; j < 4; ++j) {
            v2f b = *(const v2f*)(brow[j] + k0);
            acc[j] = wmma_f32(a, b, acc[j]);
        }
    }
#pragma unroll
    for (int j = 0; j < 4; ++j) {
        const float bias = Bv[n0 + j * 16 + ml];
#pragma unroll
        for (int r = 0; r < 8; ++r)
            Out[(m0 + r + 8 * hh) * Dn + n0 + j * 16 + ml] = acc[j][r] + bias;
    }
}

// ---------------------------------------------------------------------------
// Kernel 2: flash-style attention per (query 16-row tile, head-slice h).
// S_h = Q_h K_h^T / 8 with inner dim 16 (the nonstandard einsum contracts over
// num_heads); online softmax over rows; O_h += P @ V_h (V_h is only 16 wide,
// so the whole output accumulator is one 16x16 WMMA tile).
// P (C-layout) -> A-layout transpose goes through LDS (stride 18 floats:
// 8B-aligned b64 frag loads + conflict-free banks).
// ---------------------------------------------------------------------------
__global__ __launch_bounds__(128)
void attn_kernel(const float* __restrict__ Q, const float* __restrict__ Kmat,
                 const float* __restrict__ V, float* __restrict__ Out) {
    __shared__ float pshare[4][16 * 18];
    const int lane = threadIdx.x & 31;
    const int wave = threadIdx.x >> 5;
    const int m0 = blockIdx.x * 16;
    const int h  = blockIdx.y * 4 + wave;    // head_dim slice index, 0..63
    const int ml = lane & 15;
    const int hh = lane >> 4;
    const int col = h * NH;                  // base column of slice h

    // Q fragments (reused for every key tile): A[m,g], chunks of K=4
    v2f qf[4];
    const float* qrow = Q + (m0 + ml) * Dn + col + 2 * hh;
#pragma unroll
    for (int c = 0; c < 4; ++c) qf[c] = *(const v2f*)(qrow + 4 * c);

    float M[8], L[8];
    v8f o = {};
#pragma unroll
    for (int r = 0; r < 8; ++r) { M[r] = -3.0e38f; L[r] = 0.0f; }

    float* pbuf = pshare[wave];

    for (int n0 = 0; n0 < Tn; n0 += 16) {
        // S tile = Q_h (16x16) * K_h^T (16x16), via 4 k-chunks of 4
        v8f s = {};
        const float* krow = Kmat + (n0 + ml) * Dn + col + 2 * hh;
#pragma unroll
        for (int c = 0; c < 4; ++c) {
            v2f kf = *(const v2f*)(krow + 4 * c);
            s = wmma_f32(qf[c], kf, s);
        }
        // online softmax: rows r(+8*hh) live across the 16 lanes of each half
        float scl[8];
#pragma unroll
        for (int r = 0; r < 8; ++r) {
            float sv = s[r] * 0.125f;   // 1/sqrt(head_dim=64)
            float mx = sv;
            mx = fmaxf(mx, __shfl_xor(mx, 1, 32));
            mx = fmaxf(mx, __shfl_xor(mx, 2, 32));
            mx = fmaxf(mx, __shfl_xor(mx, 4, 32));
            mx = fmaxf(mx, __shfl_xor(mx, 8, 32));
            float mnew = fmaxf(M[r], mx);
            float p = __expf(sv - mnew);
            float sum = p;
            sum += __shfl_xor(sum, 1, 32);
            sum += __shfl_xor(sum, 2, 32);
            sum += __shfl_xor(sum, 4, 32);
            sum += __shfl_xor(sum, 8, 32);
            float sc = __expf(M[r] - mnew);
            L[r] = L[r] * sc + sum;
            M[r] = mnew;
            scl[r] = sc;
            s[r] = p;
        }
        // P tile to LDS (C layout -> row-major, stride 18)
#pragma unroll
        for (int r = 0; r < 8; ++r)
            pbuf[(r + 8 * hh) * 18 + ml] = s[r];
        asm volatile("s_wait_dscnt 0x0" ::: "memory");  // per-wave LDS fence
        // rescale running output
#pragma unroll
        for (int r = 0; r < 8; ++r) o[r] *= scl[r];
        // O += P @ V_tile (A frags from LDS, B frags from global V)
        const float* vbase = V + n0 * Dn + col + ml;
#pragma unroll
        for (int c = 0; c < 4; ++c) {
            v2f pa = *(const v2f*)(pbuf + ml * 18 + 4 * c + 2 * hh);
            v2f vb;
            vb.x = vbase[(4 * c + 2 * hh) * Dn];
            vb.y = vbase[(4 * c + 2 * hh + 1) * Dn];
            o = wmma_f32(pa, vb, o);
        }
        asm volatile("s_wait_dscnt 0x0" ::: "memory");  // drain before reuse
    }
    // normalize and store attended[t, h*16+g]
#pragma unroll
    for (int r = 0; r < 8; ++r)
        Out[(m0 + r + 8 * hh) * Dn + col + ml] = o[r] / L[r];
}

// ---------------------------------------------------------------------------
// Kernel 3: output projection. out = attended @ W_o + b_o (B NOT transposed:
// B[k,n] = W[k,n] -> two b32 loads per frag). Same 16x64 register tile.
// ---------------------------------------------------------------------------
__global__ __launch_bounds__(256)
void out_proj_kernel(const float* __restrict__ A, const float* __restrict__ W,
                     const float* __restrict__ bias, float* __restrict__ Out) {
    const int lane = threadIdx.x & 31;
    const int wave = threadIdx.x >> 5;
    const int m0 = (blockIdx.x * 2 + (wave & 1)) * 16;
    const int n0 = blockIdx.y * 256 + (wave >> 1) * 64;
    const int ml = lane & 15;
    const int hh = lane >> 4;

    const float* arow = A + (m0 + ml) * Dn + 2 * hh;
    const float* bcol[4];
#pragma unroll
    for (int j = 0; j < 4; ++j)
        bcol[j] = W + n0 + j * 16 + ml + 2 * hh * Dn;

    v8f acc[4] = {};
#pragma unroll 2
    for (int k0 = 0; k0 < Dn; k0 += 4) {
        __builtin_prefetch(arow + k0 + 128, 0, 3);
        v2f a = *(const v2f*)(arow + k0);
#pragma unroll
        for (int j = 0; j < 4; ++j) {
            v2f b;
            b.x = bcol[j][(size_t)k0 * Dn];        // W[k0+2hh,   n]
            b.y = bcol[j][(size_t)k0 * Dn + Dn];   // W[k0+2hh+1, n]
            acc[j] = wmma_f32(a, b, acc[j]);
        }
    }
#pragma unroll
    for (int j = 0; j < 4; ++j) {
        const float bv = bias[n0 + j * 16 + ml];
#pragma unroll
        for (int r = 0; r < 8; ++r)
            Out[(m0 + r + 8 * hh) * Dn + n0 + j * 16 + ml] = acc[j][r] + bv;
    }
}

// ---------------------------------------------------------------------------
extern "C" void kernel_launch(void* const* d_in, const int* in_sizes, int n_in,
                              void* d_out, int out_size, void* d_ws,
                              size_t ws_size, hipStream_t stream) {
    const float* x  = (const float*)d_in[0];
    const float* wq = (const float*)d_in[1];
    const float* bq = (const float*)d_in[2];
    const float* wk = (const float*)d_in[3];
    const float* bk = (const float*)d_in[4];
    const float* wv = (const float*)d_in[5];
    const float* bv = (const float*)d_in[6];
    const float* wo = (const float*)d_in[7];
    const float* bo = (const float*)d_in[8];
    float* out = (float*)d_out;

    // workspace: Q, K, V, attended — 4 MiB each (16 MiB total)
    float* Q  = (float*)d_ws;
    float* K  = Q + (size_t)Tn * Dn;
    float* V  = K + (size_t)Tn * Dn;
    float* At = V + (size_t)Tn * Dn;

    dim3 gproj(Tn / 32, Dn / 256, 3);
    qkv_proj_kernel<<<gproj, 256, 0, stream>>>(x, wq, bq, wk, bk, wv, bv,
                                               Q, K, V);

    dim3 gattn(Tn / 16, HD / 4);   // 64 query tiles x 16 blocks of 4 h-slices
    attn_kernel<<<gattn, 128, 0, stream>>>(Q, K, V, At);

    dim3 gout(Tn / 32, Dn / 256);
    out_proj_kernel<<<gout, 256, 0, stream>>>(At, wo, bo, out);
}